// NERHead_85401129714008
// MI455X (gfx1250) — compile-verified
//
#include <hip/hip_runtime.h>
#include <hip/hip_bf16.h>
#include <math.h>

// Problem constants: B=64, S=512, H=768, T=9 (padded to 16 in emissions).
#define NB 64
#define NS 512
#define NH 768
#define NT 9
#define NTP 16

typedef float v2f __attribute__((ext_vector_type(2)));
typedef float v8f __attribute__((ext_vector_type(8)));

// ---------------------------------------------------------------------------
// Kernel 1: emissions = relu(emb @ W^T + b), via V_WMMA_F32_16X16X4_F32.
// Grid: 256 blocks x 256 threads (8 waves). Each wave computes one 16-row
// M-tile of the flattened (B*S, 16) output: 2048 tiles total.
// A-frag (16x4 f32): lane l -> row = l&15, k-pair = 2*(l>>4).
// B-frag (4x16 f32): lane l -> col n = l&15 (W row n, clamped; cols >=9 are
// pad, never read downstream), same k-pair mapping.
// C/D (16x16 f32, 8 VGPRs): lane l, reg r -> M = r + 8*(l>>4), N = l&15.
// ---------------------------------------------------------------------------
__global__ __launch_bounds__(256) void ner_emis_gemm(
    const float* __restrict__ emb,   // [B*S, H]
    const float* __restrict__ Wg,    // [T, H]
    const float* __restrict__ bg,    // [T]
    float* __restrict__ em_out)      // [B*S, NTP]
{
    const int lane = threadIdx.x & 31;
    const int wv   = threadIdx.x >> 5;
    const int mt   = blockIdx.x * 8 + wv;     // M-tile index, 0..2047
    const int s0   = mt << 4;                 // base flattened row
    const int nrow = lane & 15;
    const int half = lane >> 4;
    const int kb   = half << 1;               // 0 or 2

    const float* ap = emb + (size_t)(s0 + nrow) * NH + kb;
    const float* wp = Wg  + (size_t)((nrow < NT) ? nrow : 0) * NH + kb;

    v8f acc = {0.f, 0.f, 0.f, 0.f, 0.f, 0.f, 0.f, 0.f};

    for (int k0 = 0; k0 < NH; k0 += 4) {
        if ((k0 & 31) == 0) {
            __builtin_prefetch(ap + k0 + 512, 0, 3);   // global_prefetch_b8
        }
        v2f av = *(const v2f*)(ap + k0);
        v2f bv = *(const v2f*)(wp + k0);
        // D = A(16x4) * B(4x16) + C ; 8 args: neg_a, A, neg_b, B, c_mod, C,
        // reuse_a, reuse_b
        acc = __builtin_amdgcn_wmma_f32_16x16x4_f32(
            false, av, false, bv, (short)0, acc, false, false);
    }

    const float bias = (nrow < NT) ? bg[nrow] : 0.0f;
#pragma unroll
    for (int r = 0; r < 8; ++r) {
        float v = acc[r] + bias;
        v = fmaxf(v, 0.0f);                       // relu
        em_out[(size_t)(s0 + r + 8 * half) * NTP + nrow] = v;
    }
}

// ---------------------------------------------------------------------------
// Kernel 2: CRF negative log-likelihood, one wave32 per batch element.
// Lanes 0..8 carry alpha[j]; __shfl gathers alpha across lanes each step.
// Emissions/labels/mask/transitions staged in LDS; recursion is LDS-only.
// ---------------------------------------------------------------------------
__global__ __launch_bounds__(32) void ner_crf(
    const float* __restrict__ em,    // [B*S, NTP] from kernel 1
    const float* __restrict__ st,    // start_trans [T]
    const float* __restrict__ trg,   // transitions [T,T]
    const float* __restrict__ en,    // end_trans [T]
    const int*   __restrict__ lab,   // labels [B,S]
    const int*   __restrict__ msk,   // mask [B,S]
    float* __restrict__ nll)         // [B]
{
    __shared__ float lds_em[NS * NTP];   // 32 KB
    __shared__ float lds_tr[NT * NT];
    __shared__ int   lds_lab[NS];
    __shared__ int   lds_msk[NS];

    const int b = blockIdx.x;
    const int j = threadIdx.x;           // lane 0..31

    // Stage emissions for this batch (2048 float4 loads across 32 lanes).
    const float4* src = (const float4*)(em + (size_t)b * NS * NTP);
    float4* dst = (float4*)lds_em;
    for (int i = j; i < NS * NTP / 4; i += 32) dst[i] = src[i];
    for (int i = j; i < NS; i += 32) {
        lds_lab[i] = lab[b * NS + i];
        lds_msk[i] = msk[b * NS + i];
    }
    for (int i = j; i < NT * NT; i += 32) lds_tr[i] = trg[i];
    __syncthreads();

    // Per-lane transition column: tr[i] = transitions[i][j].
    float tr[NT];
#pragma unroll
    for (int i = 0; i < NT; ++i)
        tr[i] = (j < NT) ? lds_tr[i * NT + j] : -1e30f;

    const int jl = (j < NTP) ? j : 0;    // clamp for padded-lane LDS reads

    // alpha_0 = start + emissions[0]
    float alpha = (j < NT) ? (st[j] + lds_em[jl]) : -1e30f;

    // Numerator state (meaningful on lane 0).
    int lraw0 = lds_lab[0];
    int lab0  = (lraw0 == -100) ? 0 : lraw0;
    float num = st[lab0] + lds_em[lab0];
    bool m0   = (lds_msk[0] != 0) && (lraw0 != -100);
    int last_lab = m0 ? lab0 : 0;
    int prev     = lab0;

    for (int t = 1; t < NS; ++t) {
        float v[NT];
        float mx = -3.0e38f;
#pragma unroll
        for (int i = 0; i < NT; ++i) {
            float ai = __shfl(alpha, i, 32);
            v[i] = ai + tr[i];
            mx = fmaxf(mx, v[i]);
        }
        float ss = 0.0f;
#pragma unroll
        for (int i = 0; i < NT; ++i) ss += __expf(v[i] - mx);

        float em_tj = lds_em[t * NTP + jl];
        float na = mx + __logf(ss) + em_tj;

        int  lraw = lds_lab[t];
        bool mv   = (lds_msk[t] != 0) && (lraw != -100);
        alpha = mv ? na : alpha;

        int lmod = (lraw == -100) ? 0 : lraw;
        if (j == 0) {
            if (mv) {
                num += lds_em[t * NTP + lmod] + lds_tr[prev * NT + lmod];
                last_lab = lmod;
            }
            prev = lmod;   // chain uses labels_mod[t-1] unconditionally
        }
    }

    // denom = logsumexp_j(alpha_j + end_trans_j)
    float aend = (j < NT) ? (alpha + en[j]) : -3.0e38f;
    float mx = -3.0e38f;
    float g[NT];
#pragma unroll
    for (int i = 0; i < NT; ++i) {
        g[i] = __shfl(aend, i, 32);
        mx = fmaxf(mx, g[i]);
    }
    float ss = 0.0f;
#pragma unroll
    for (int i = 0; i < NT; ++i) ss += __expf(g[i] - mx);

    if (j == 0) {
        float denom = mx + __logf(ss);
        nll[b] = denom - (num + en[last_lab]);
    }
}

// ---------------------------------------------------------------------------
// Kernel 3: deterministic fixed-order mean of 64 per-batch NLLs -> d_out[0].
// ---------------------------------------------------------------------------
__global__ void ner_reduce(const float* __restrict__ nll,
                           float* __restrict__ out)
{
    if (threadIdx.x == 0 && blockIdx.x == 0) {
        float s = 0.0f;
        for (int i = 0; i < NB; ++i) s += nll[i];
        out[0] = s * (1.0f / (float)NB);
    }
}

extern "C" void kernel_launch(void* const* d_in, const int* in_sizes, int n_in,
                              void* d_out, int out_size, void* d_ws, size_t ws_size,
                              hipStream_t stream) {
    (void)in_sizes; (void)n_in; (void)out_size; (void)ws_size;
    const float* emb  = (const float*)d_in[0];   // [B,S,H] f32
    const float* Wg   = (const float*)d_in[1];   // [T,H]   f32
    const float* bg   = (const float*)d_in[2];   // [T]     f32
    const float* stt  = (const float*)d_in[3];   // [T]     f32
    const float* trg  = (const float*)d_in[4];   // [T,T]   f32
    const float* ent  = (const float*)d_in[5];   // [T]     f32
    const int*   lab  = (const int*)d_in[6];     // [B,S]   i32
    const int*   msk  = (const int*)d_in[7];     // [B,S]   i32

    float* em_ws  = (float*)d_ws;                        // 32768*16 f32 = 2 MB
    float* nll_ws = em_ws + (size_t)NB * NS * NTP;       // 64 f32

    ner_emis_gemm<<<256, 256, 0, stream>>>(emb, Wg, bg, em_ws);
    ner_crf<<<NB, 32, 0, stream>>>(em_ws, stt, trg, ent, lab, msk, nll_ws);
    ner_reduce<<<1, 32, 0, stream>>>(nll_ws, (float*)d_out);
}